// ActivationInterface_56521769616082
// MI455X (gfx1250) — compile-verified
//
#include <hip/hip_runtime.h>
#include <stdint.h>

// ---------------------------------------------------------------------------
// Gumbel-softmax tabular activation for MI455X (gfx1250, wave32).
//   out[:, :64]   = tanh(x[:, :64])
//   out[:, 64:]   = one_hot(argmax_k((logit_k + g1_k)/tau + g2_k)) per feature
// (categorical sampling of log-softmax collapses to a single Gumbel argmax)
//
// HBM-bound: ~671 MB total traffic -> ~29 us at 23.3 TB/s. CDNA5 paths used:
//   * global_load_async_to_lds_b128 (ASYNCcnt) to stage each row into LDS
//   * s_wait_asynccnt 0 before LDS consumption
//   * non-temporal b128 stores (output stream > 192 MB L2)
//   * v_tanh_f32 via __builtin_amdgcn_tanhf when available
// ---------------------------------------------------------------------------

#define B_ROWS          262144
#define NUM_COLS        64
#define NCAT            16
#define KCAT            16
#define COLS            (NUM_COLS + NCAT * KCAT)   // 320 floats = 1280 B/row
#define ROWS_PER_BLOCK  8
#define THREADS         256                        // 8 wave32 per block
#define INV_TAU         5.0f                       // 1 / 0.2
#define TOL             1e-20f

typedef float v4f __attribute__((ext_vector_type(4)));

__device__ __forceinline__ uint32_t hash_u32(uint32_t x) {
  x ^= x >> 16; x *= 0x7feb352du;
  x ^= x >> 15; x *= 0x846ca68bu;
  x ^= x >> 16;
  return x;
}

__device__ __forceinline__ float u01(uint32_t h) {
  return (float)(h >> 8) * 0x1p-24f;   // [0, 1)
}

// g = -log(-log(u + TOL) + TOL) ; two v_log_f32 trans ops
__device__ __forceinline__ float gumbel_from_u(float u) {
  float inner = -__logf(u + TOL);
  return -__logf(inner + TOL);
}

__device__ __forceinline__ float fast_tanh(float x) {
#if __has_builtin(__builtin_amdgcn_tanhf)
  return __builtin_amdgcn_tanhf(x);          // v_tanh_f32 (CDNA5 trans op)
#else
  float e = __expf(-2.0f * fabsf(x));
  float r = (1.0f - e) / (1.0f + e);
  return copysignf(r, x);
#endif
}

__global__ __launch_bounds__(THREADS)
void ActivationInterface_gumbel_kernel(const float* __restrict__ x,
                                       float* __restrict__ out) {
  __shared__ float smem[ROWS_PER_BLOCK * COLS];  // 10240 B per block

  const int lane = threadIdx.x & 31;
  const int wave = threadIdx.x >> 5;
  const int row  = blockIdx.x * ROWS_PER_BLOCK + wave;

  // ---- async-stage this wave's 1280 B row into its LDS slice ---------------
  // 3 x b128 per lane-group: 32 lanes * 16 B = 512 B per issue, third issue
  // predicated to lanes 0..15 (256 B). Tracked on ASYNCcnt, no VGPR staging.
  const uint64_t grow = (uint64_t)(uintptr_t)(x + (size_t)row * COLS);
  const uint32_t lrow = (uint32_t)(uintptr_t)(&smem[wave * COLS]); // low 32b = LDS addr

  uint64_t g0 = grow + 16u * (uint32_t)lane;
  uint32_t l0 = lrow + 16u * (uint32_t)lane;
  asm volatile("global_load_async_to_lds_b128 %0, %1, off"
               :: "v"(l0), "v"(g0) : "memory");
  uint64_t g1 = g0 + 512u;
  uint32_t l1 = l0 + 512u;
  asm volatile("global_load_async_to_lds_b128 %0, %1, off"
               :: "v"(l1), "v"(g1) : "memory");
  if (lane < 16) {
    uint64_t g2 = g0 + 1024u;
    uint32_t l2 = l0 + 1024u;
    asm volatile("global_load_async_to_lds_b128 %0, %1, off"
                 :: "v"(l2), "v"(g2) : "memory");
  }
  asm volatile("s_wait_asynccnt 0x0" ::: "memory");

  const float* rsm  = &smem[wave * COLS];
  float*       orow = out + (size_t)row * COLS;

  if (lane < 16) {
    // ---- numerical block: lane l -> tanh of float4 at cols [4l, 4l+3] -----
    v4f v = *reinterpret_cast<const v4f*>(rsm + 4 * lane);
    v4f o;
    o.x = fast_tanh(v.x);
    o.y = fast_tanh(v.y);
    o.z = fast_tanh(v.z);
    o.w = fast_tanh(v.w);
    __builtin_nontemporal_store(o, reinterpret_cast<v4f*>(orow + 4 * lane));
  } else {
    // ---- categorical block: lane owns feature f (16 contiguous logits) ----
    const int f = lane - 16;
    const float* lg = rsm + NUM_COLS + KCAT * f;
    const uint32_t ebase = (uint32_t)row * (uint32_t)(NCAT * KCAT)
                         + (uint32_t)(f * KCAT);
    int   best  = 0;
    float bestv = -3.4e38f;
#pragma unroll
    for (int k = 0; k < KCAT; ++k) {
      const uint32_t e = ebase + (uint32_t)k;
      const float gA = gumbel_from_u(u01(hash_u32(e ^ 0x9E3779B9u))); // noise stream
      const float gB = gumbel_from_u(u01(hash_u32(e ^ 0x85EBCA6Bu))); // sample stream
      const float s  = (lg[k] + gA) * INV_TAU + gB;   // Gumbel-max == categorical(log soft)
      if (s > bestv) { bestv = s; best = k; }
    }
    float* op = orow + NUM_COLS + KCAT * f;
#pragma unroll
    for (int q = 0; q < 4; ++q) {
      const int b0 = 4 * q;
      v4f o;
      o.x = (best == b0 + 0) ? 1.0f : 0.0f;
      o.y = (best == b0 + 1) ? 1.0f : 0.0f;
      o.z = (best == b0 + 2) ? 1.0f : 0.0f;
      o.w = (best == b0 + 3) ? 1.0f : 0.0f;
      __builtin_nontemporal_store(o, reinterpret_cast<v4f*>(op + b0));
    }
  }
}

extern "C" void kernel_launch(void* const* d_in, const int* in_sizes, int n_in,
                              void* d_out, int out_size, void* d_ws, size_t ws_size,
                              hipStream_t stream) {
  (void)in_sizes; (void)n_in; (void)out_size; (void)d_ws; (void)ws_size;
  const float* x   = (const float*)d_in[0];
  float*       out = (float*)d_out;

  dim3 grid(B_ROWS / ROWS_PER_BLOCK);   // 32768 blocks
  dim3 block(THREADS);                  // 8 wave32 per block, 1 row per wave
  ActivationInterface_gumbel_kernel<<<grid, block, 0, stream>>>(x, out);
}